// AttentionCrop_84224308675206
// MI455X (gfx1250) — compile-verified
//
#include <hip/hip_runtime.h>
#include <math.h>

// sigmoid sharpness constant from the reference
#define KK_SHARP 10.0f

typedef float v4f __attribute__((ext_vector_type(4)));

__device__ __forceinline__ float sigk(float x) {
    // sigmoid(KK * x)
    return 1.0f / (1.0f + __expf(-KK_SHARP * x));
}

// Per-row boundary math, exactly matching the reference semantics.
__device__ __forceinline__ void row_bounds(float tb, float lb, float rsum,
                                           int rfirst, int rlast, int L,
                                           float& tl_out, float& tr_out) {
    // argmax(mask) == first nonzero (or 0 if all-zero);
    // L - argmax(mask[::-1]) == last nonzero + 1 (or L if all-zero)
    float left  = (rlast < 0) ? -1.0f : (float)rfirst - 1.0f;
    float right = (rlast < 0) ? (float)L : (float)(rlast + 1);
    float sl    = rsum * 0.5f;                  // sentence_length
    float leff  = (lb <= sl) ? sl : lb;         // where(l <= sl, sl, l)

    float a  = tb - leff;
    float tl = (a >= left) ? a : 0.0f;          // (t-l)*(t-l >= left)
    if (tl == 0.0f) tl = left;                  // where(t_left==0, left, .)

    float bb = tb + leff;
    float tr = (bb <= right) ? bb : 0.0f;       // (t+l <= right)*(t+l)
    if (tr == 0.0f) tr = right;                 // where(t_right==0, right, .)

    tl_out = tl;
    tr_out = tr;
}

// ---------------------------------------------------------------------------
// Specialized kernel: L == 4096 (the reference shape).
// 256 threads = 8 wave32 waves; each thread owns 4 float4 chunks, strided by
// 4 KB. The async-to-LDS op adds its immediate offset to BOTH the LDS and the
// global address, so one base pair + 4 immediates needs zero address VALU.
// ---------------------------------------------------------------------------
__global__ __launch_bounds__(256) void attention_crop_4096(
    const float* __restrict__ t_in,
    const float* __restrict__ l_in,
    const float* __restrict__ mask,
    float* __restrict__ out)
{
    constexpr int L = 4096;
    __shared__ __align__(16) float smem[L];     // staged mask row (16 KB)
    __shared__ float s_sum[8];
    __shared__ int   s_first[8];
    __shared__ int   s_last[8];
    __shared__ float s_tl;
    __shared__ float s_tr;

    const int b   = blockIdx.x;
    const int tid = threadIdx.x;
    const float* __restrict__ row  = mask + (size_t)b * (size_t)L;
    float* __restrict__       orow = out  + (size_t)b * (size_t)L;

    // ---- Phase 1: async global -> LDS staging (ASYNCcnt path, NT hint) ----
    {
        unsigned lds_off = (unsigned)(size_t)(void*)smem + ((unsigned)tid << 4);
        const float* g = row + (tid << 2);
#define ASYNC_LD(OFF)                                                          \
        asm volatile("global_load_async_to_lds_b128 %0, %1, off offset:" #OFF \
                     " th:TH_LOAD_NT"                                          \
                     :: "v"(lds_off), "v"(g) : "memory")
        ASYNC_LD(0);
        ASYNC_LD(4096);
        ASYNC_LD(8192);
        ASYNC_LD(12288);
#undef ASYNC_LD
    }
    asm volatile("s_wait_asynccnt 0" ::: "memory");
    __syncthreads();

    // ---- Phase 2: row reduction: sum, first nonzero, last nonzero ----
    float psum   = 0.0f;
    int   pfirst = 0x7fffffff;
    int   plast  = -1;
    #pragma unroll
    for (int k = 0; k < 4; ++k) {
        int c = tid + (k << 8);
        v4f v = ((const v4f*)smem)[c];
        int j = c << 2;
        psum += (v.x + v.y) + (v.z + v.w);
        if (v.x != 0.0f) { pfirst = min(pfirst, j    ); plast = max(plast, j    ); }
        if (v.y != 0.0f) { pfirst = min(pfirst, j + 1); plast = max(plast, j + 1); }
        if (v.z != 0.0f) { pfirst = min(pfirst, j + 2); plast = max(plast, j + 2); }
        if (v.w != 0.0f) { pfirst = min(pfirst, j + 3); plast = max(plast, j + 3); }
    }

    // wave32 butterfly reduction (gfx1250 is wave32-only)
    #pragma unroll
    for (int off = 16; off > 0; off >>= 1) {
        psum   += __shfl_xor(psum,   off, 32);
        pfirst  = min(pfirst, __shfl_xor(pfirst, off, 32));
        plast   = max(plast,  __shfl_xor(plast,  off, 32));
    }
    const int wid  = tid >> 5;
    const int lane = tid & 31;
    if (lane == 0) { s_sum[wid] = psum; s_first[wid] = pfirst; s_last[wid] = plast; }
    __syncthreads();

    // ---- Phase 3: per-row scalar boundary math ----
    if (tid == 0) {
        float rsum = 0.0f; int rfirst = 0x7fffffff; int rlast = -1;
        #pragma unroll
        for (int w = 0; w < 8; ++w) {
            rsum  += s_sum[w];
            rfirst = min(rfirst, s_first[w]);
            rlast  = max(rlast,  s_last[w]);
        }
        row_bounds(t_in[b], l_in[b], rsum, rfirst, rlast, L, s_tl, s_tr);
    }
    __syncthreads();

    const float tl = s_tl;
    const float tr = s_tr;

    // ---- Phase 4: elementwise crop mask, nontemporal b128 stores ----
    #pragma unroll
    for (int k = 0; k < 4; ++k) {
        int c = tid + (k << 8);
        v4f v = ((const v4f*)smem)[c];
        float p = (float)(c << 2);
        v4f o;
        float cm;
        cm = sigk(p        - tl) - sigk(p        - tr); o.x = (cm >= 0.5f) ? v.x : 0.0f;
        cm = sigk(p + 1.0f - tl) - sigk(p + 1.0f - tr); o.y = (cm >= 0.5f) ? v.y : 0.0f;
        cm = sigk(p + 2.0f - tl) - sigk(p + 2.0f - tr); o.z = (cm >= 0.5f) ? v.z : 0.0f;
        cm = sigk(p + 3.0f - tl) - sigk(p + 3.0f - tr); o.w = (cm >= 0.5f) ? v.w : 0.0f;
        __builtin_nontemporal_store(o, (v4f*)orow + c);
    }
}

// ---------------------------------------------------------------------------
// Generic fallback for arbitrary L (same algorithm, runtime loop bounds).
// ---------------------------------------------------------------------------
__global__ __launch_bounds__(256) void attention_crop_generic(
    const float* __restrict__ t_in,
    const float* __restrict__ l_in,
    const float* __restrict__ mask,
    float* __restrict__ out,
    int L)
{
    extern __shared__ __align__(16) float smem[];   // L floats
    __shared__ float s_sum[8];
    __shared__ int   s_first[8];
    __shared__ int   s_last[8];
    __shared__ float s_tl;
    __shared__ float s_tr;

    const int b   = blockIdx.x;
    const int tid = threadIdx.x;
    const int L4  = L >> 2;
    const float* __restrict__ row  = mask + (size_t)b * (size_t)L;
    float* __restrict__       orow = out  + (size_t)b * (size_t)L;

    unsigned lds_base = (unsigned)(size_t)(void*)smem;
    for (int c = tid; c < L4; c += 256) {
        unsigned lds_off = lds_base + ((unsigned)c << 4);
        const float* g = row + (c << 2);
        asm volatile("global_load_async_to_lds_b128 %0, %1, off"
                     :: "v"(lds_off), "v"(g) : "memory");
    }
    for (int j = (L4 << 2) + tid; j < L; j += 256) smem[j] = row[j];
    asm volatile("s_wait_asynccnt 0" ::: "memory");
    __syncthreads();

    float psum   = 0.0f;
    int   pfirst = 0x7fffffff;
    int   plast  = -1;
    for (int c = tid; c < L4; c += 256) {
        v4f v = ((const v4f*)smem)[c];
        int j = c << 2;
        psum += (v.x + v.y) + (v.z + v.w);
        if (v.x != 0.0f) { pfirst = min(pfirst, j    ); plast = max(plast, j    ); }
        if (v.y != 0.0f) { pfirst = min(pfirst, j + 1); plast = max(plast, j + 1); }
        if (v.z != 0.0f) { pfirst = min(pfirst, j + 2); plast = max(plast, j + 2); }
        if (v.w != 0.0f) { pfirst = min(pfirst, j + 3); plast = max(plast, j + 3); }
    }
    for (int j = (L4 << 2) + tid; j < L; j += 256) {
        float v = smem[j];
        psum += v;
        if (v != 0.0f) { pfirst = min(pfirst, j); plast = max(plast, j); }
    }

    #pragma unroll
    for (int off = 16; off > 0; off >>= 1) {
        psum   += __shfl_xor(psum,   off, 32);
        pfirst  = min(pfirst, __shfl_xor(pfirst, off, 32));
        plast   = max(plast,  __shfl_xor(plast,  off, 32));
    }
    const int wid  = tid >> 5;
    const int lane = tid & 31;
    if (lane == 0) { s_sum[wid] = psum; s_first[wid] = pfirst; s_last[wid] = plast; }
    __syncthreads();

    if (tid == 0) {
        float rsum = 0.0f; int rfirst = 0x7fffffff; int rlast = -1;
        #pragma unroll
        for (int w = 0; w < 8; ++w) {
            rsum  += s_sum[w];
            rfirst = min(rfirst, s_first[w]);
            rlast  = max(rlast,  s_last[w]);
        }
        row_bounds(t_in[b], l_in[b], rsum, rfirst, rlast, L, s_tl, s_tr);
    }
    __syncthreads();

    const float tl = s_tl;
    const float tr = s_tr;

    for (int c = tid; c < L4; c += 256) {
        v4f v = ((const v4f*)smem)[c];
        float p = (float)(c << 2);
        v4f o;
        float cm;
        cm = sigk(p        - tl) - sigk(p        - tr); o.x = (cm >= 0.5f) ? v.x : 0.0f;
        cm = sigk(p + 1.0f - tl) - sigk(p + 1.0f - tr); o.y = (cm >= 0.5f) ? v.y : 0.0f;
        cm = sigk(p + 2.0f - tl) - sigk(p + 2.0f - tr); o.z = (cm >= 0.5f) ? v.z : 0.0f;
        cm = sigk(p + 3.0f - tl) - sigk(p + 3.0f - tr); o.w = (cm >= 0.5f) ? v.w : 0.0f;
        __builtin_nontemporal_store(o, (v4f*)orow + c);
    }
    for (int j = (L4 << 2) + tid; j < L; j += 256) {
        float v = smem[j];
        float p = (float)j;
        float cm = sigk(p - tl) - sigk(p - tr);
        orow[j] = (cm >= 0.5f) ? v : 0.0f;
    }
}

extern "C" void kernel_launch(void* const* d_in, const int* in_sizes, int n_in,
                              void* d_out, int out_size, void* d_ws, size_t ws_size,
                              hipStream_t stream) {
    (void)n_in; (void)d_ws; (void)ws_size; (void)out_size;
    const float* t    = (const float*)d_in[0];   // [B,1]
    const float* l    = (const float*)d_in[1];   // [B,1]
    const float* mask = (const float*)d_in[2];   // [B,L]
    const int B = in_sizes[0];
    const int L = (B > 0) ? (in_sizes[2] / B) : 0;
    if (B <= 0 || L <= 0) return;

    if (L == 4096) {
        attention_crop_4096<<<B, 256, 0, stream>>>(t, l, mask, (float*)d_out);
    } else {
        const size_t smem_bytes = (size_t)L * sizeof(float);
        attention_crop_generic<<<B, 256, smem_bytes, stream>>>(
            t, l, mask, (float*)d_out, L);
    }
}